// Net_1975684956439
// MI455X (gfx1250) — compile-verified
//
#include <hip/hip_runtime.h>

// ---------------------------------------------------------------------------
// Problem constants (from the reference)
// ---------------------------------------------------------------------------
#define NUM_ATOMS 2048
#define NUM_DESC  64
#define NUM_PAIRS (NUM_ATOMS * 64)          // 131072
#define NUM_ROWS  (3 * NUM_PAIRS)           // flattened [3*P, 64] row count
#define OUT_ELEMS (1 + 3 * NUM_ATOMS)       // [e_pa, f(3,2048)]

typedef __attribute__((ext_vector_type(2))) float v2f;
typedef __attribute__((ext_vector_type(4))) float v4f;
typedef __attribute__((ext_vector_type(8))) float v8f;

// Each wave handles GROUPS_PER_WAVE tiles of 16 rows (16x64 f32 = 4 KB each).
#define GROUPS_PER_WAVE 4
#define ROWS_PER_WAVE   (16 * GROUPS_PER_WAVE)

// ---------------------------------------------------------------------------
// Kernel 1: initialize output (harness poisons d_out with 0xAA).
// out[0] = bias, out[1..] = 0 (scatter targets).
// ---------------------------------------------------------------------------
__global__ void init_out_kernel(float* __restrict__ out,
                                const float* __restrict__ bias) {
    int i = blockIdx.x * blockDim.x + threadIdx.x;
    if (i < OUT_ELEMS) out[i] = (i == 0) ? bias[0] : 0.0f;
}

// ---------------------------------------------------------------------------
// Kernel 2: e_pa = sum(coeffs * w) / NUM_ATOMS  (accumulated into out[0]).
// 512 KB read once -> non-temporal b128 loads; grid-stride, wave32 shuffle
// reduce, few hundred atomics total.
// ---------------------------------------------------------------------------
__global__ __launch_bounds__(256) void energy_kernel(
    const float* __restrict__ coeffs,   // [NUM_ATOMS * NUM_DESC]
    const float* __restrict__ w,        // [NUM_DESC]
    float* __restrict__ out) {
    const int tid    = blockIdx.x * blockDim.x + threadIdx.x;
    const int stride = gridDim.x * blockDim.x;
    const int total4 = (NUM_ATOMS * NUM_DESC) / 4;

    float s = 0.0f;
    for (int i = tid; i < total4; i += stride) {
        v4f c4 = __builtin_nontemporal_load((const v4f*)coeffs + i);
        int d = (i * 4) & (NUM_DESC - 1);   // descriptor index is i*4 mod 64
        s += c4.x * w[d + 0] + c4.y * w[d + 1] +
             c4.z * w[d + 2] + c4.w * w[d + 3];
    }
    // wave32 reduction
    #pragma unroll
    for (int off = 16; off > 0; off >>= 1)
        s += __shfl_down(s, off, 32);
    if ((threadIdx.x & 31) == 0)
        atomicAdd(out, s * (1.0f / (float)NUM_ATOMS));
}

// ---------------------------------------------------------------------------
// Kernel 3: force contraction + scatter (the 100 MB, bandwidth-bound part).
//
// derivs viewed as R = 3*P rows of 64 f32.  Per 16-row tile, a wave runs
// 16 V_WMMA_F32_16X16X4_F32 ops split over TWO accumulators (even/odd K
// chunk) to halve the D->C RAW chain, then one v8f add merges them.
//   A layout (ISA 7.12.2, 32-bit A 16x4): lane m (0-15) = row m, VGPR0/1
//     hold K=0/1;  lane 16+m = row m, VGPR0/1 hold K=2/3.
//     -> lane loads 2 consecutive f32 at row*64 + 4*kc + 2*half (b64 load),
//        marked non-temporal (single-use 100 MB stream, keep L2 for the
//        scatter targets which are hit ~64x each).
//   B: all 16 columns = weight chunk, so every column of D is the dot.
//     Mirrors A's half-lane K split: b = (w[4kc+2h], w[4kc+2h+1]).
//   C/D layout: VGPR v = row v (lanes 0-15) / row 8+v (lanes 16-31).
//     -> lanes 0-7 extract rows 0-7, lanes 16-23 extract rows 8-15.
// Scatter: out[1 + c*2048 + neigh[p]] -= dot  via global_atomic_add_f32.
// All WMMAs execute with EXEC all-1s (grid exactly covers NUM_ROWS; lane
// predication only happens in the scatter tail after the matrix ops).
// ---------------------------------------------------------------------------
__global__ __launch_bounds__(256) void force_wmma_kernel(
    const float* __restrict__ derivs,   // [3*P, 64] contiguous
    const int*   __restrict__ neigh,    // [P]
    const float* __restrict__ w,        // [64]
    float* __restrict__ out) {
    const int lane = threadIdx.x & 31;
    const int wave = (blockIdx.x * blockDim.x + threadIdx.x) >> 5;
    const int half = lane >> 4;         // 0: K0..1, 1: K2..3
    const int m    = lane & 15;         // A-matrix row within tile

    // Preload B operand for all 16 K-chunks (32 VGPRs; hits L0/L2, tiny).
    v2f bw[16];
    #pragma unroll
    for (int kc = 0; kc < 16; ++kc) {
        bw[kc].x = w[kc * 4 + half * 2 + 0];
        bw[kc].y = w[kc * 4 + half * 2 + 1];
    }

    const int base_row = wave * ROWS_PER_WAVE;
    #pragma unroll
    for (int g = 0; g < GROUPS_PER_WAVE; ++g) {
        const int row0 = base_row + g * 16;
        const float* ap =
            derivs + (size_t)(row0 + m) * NUM_DESC + (size_t)(half * 2);

        v8f acc0 = {};
        v8f acc1 = {};
        #pragma unroll
        for (int kc = 0; kc < 16; kc += 2) {
            v2f a0 = __builtin_nontemporal_load((const v2f*)(ap + kc * 4));
            v2f a1 = __builtin_nontemporal_load((const v2f*)(ap + kc * 4 + 4));
            // 8 args: (neg_a, A, neg_b, B, c_mod, C, reuse_a, reuse_b)
            acc0 = __builtin_amdgcn_wmma_f32_16x16x4_f32(
                false, a0, false, bw[kc + 0], (short)0, acc0, false, false);
            acc1 = __builtin_amdgcn_wmma_f32_16x16x4_f32(
                false, a1, false, bw[kc + 1], (short)0, acc1, false, false);
        }
        v8f acc = acc0 + acc1;

        // Extract one row-dot per active lane (all D columns identical).
        float dot = 0.0f;
        #pragma unroll
        for (int v = 0; v < 8; ++v)
            if ((lane & 7) == v) dot = acc[v];

        if ((lane & 8) == 0) {              // lanes 0-7 and 16-23 write
            int r = row0 + (lane & 7) + half * 8;   // global flattened row
            int c = r / NUM_PAIRS;                  // force component 0..2
            int p = r - c * NUM_PAIRS;              // pair index
            int n = neigh[p];
            atomicAdd(out + 1 + c * NUM_ATOMS + n, -dot);
        }
    }
}

// ---------------------------------------------------------------------------
// Host-side launcher.  Inputs (flat):
//   0: coeffs              [1,2048,64]   f32
//   1: coeffs_derivs       [1,3,P,64]    f32
//   2: central_atom_index  [P]           i32   (unused by the reference math)
//   3: neigh_atom_index    [P]           i32
//   4: weight1             [1,64]        f32
//   5: bias1               [1]           f32
// Output: [e_pa (1), out_f (3*2048)] f32, concatenated.
// ---------------------------------------------------------------------------
extern "C" void kernel_launch(void* const* d_in, const int* in_sizes, int n_in,
                              void* d_out, int out_size, void* d_ws,
                              size_t ws_size, hipStream_t stream) {
    (void)in_sizes; (void)n_in; (void)out_size; (void)d_ws; (void)ws_size;

    const float* coeffs = (const float*)d_in[0];
    const float* derivs = (const float*)d_in[1];
    const int*   neigh  = (const int*)d_in[3];
    const float* w      = (const float*)d_in[4];
    const float* bias   = (const float*)d_in[5];
    float* out = (float*)d_out;

    // 1) init output (bias + zeroed force accumulators)
    init_out_kernel<<<(OUT_ELEMS + 255) / 256, 256, 0, stream>>>(out, bias);

    // 2) energy reduction (512 KB)
    energy_kernel<<<64, 256, 0, stream>>>(coeffs, w, out);

    // 3) WMMA force contraction + scatter (100 MB, bandwidth bound)
    //    NUM_ROWS / ROWS_PER_WAVE waves, 8 waves (256 threads) per block.
    const int n_waves  = NUM_ROWS / ROWS_PER_WAVE;   // 6144
    const int n_blocks = n_waves / 8;                // 768
    force_wmma_kernel<<<n_blocks, 256, 0, stream>>>(derivs, neigh, w, out);
}